// SegmentEmbed_47425028882618
// MI455X (gfx1250) — compile-verified
//
#include <hip/hip_runtime.h>
#include <hip/hip_bf16.h>

typedef __attribute__((ext_vector_type(2))) float v2f;
typedef __attribute__((ext_vector_type(8))) float v8f;

#define B_   16
#define H_   224
#define W_   224
#define HW_  (H_ * W_)      // 50176
#define CIN_ 3
#define CI_  64             // inter_dim
#define CE_  768            // embed_dim
#define S_   196            // NUM_SEG
#define KK_  147            // 3 * 7 * 7
#define KKP_ 148            // padded: multiple of 4, 148 % 64 == 20 -> bank-clean
#define K2_  192            // 3 * 64 (conv2 im2col K)
#define K2P_ 196            // padded row: 196 % 64 == 4 -> conflict-free b64 frags
#define MB_  128            // conv1 pixel tile per block

// ---------------------------------------------------------------------------
// Kernel 1: per-batch segment-id remap (unique-inverse), counts, f_segment out
// ---------------------------------------------------------------------------
__global__ void seg_remap_kernel(const int* __restrict__ seg,
                                 int* __restrict__ fseg_int,   // [B][HW]
                                 float* __restrict__ fseg_out, // [B][HW] (output 2)
                                 int* __restrict__ cnt) {      // [B][S]
    __shared__ int present[S_];
    __shared__ int rank[S_];
    __shared__ int lcnt[S_];
    const int b   = blockIdx.x;
    const int tid = threadIdx.x;
    for (int i = tid; i < S_; i += blockDim.x) { present[i] = 0; lcnt[i] = 0; }
    __syncthreads();
    const int* segb = seg + (size_t)b * HW_;
    for (int p = tid; p < HW_; p += blockDim.x) present[segb[p]] = 1;
    __syncthreads();
    if (tid == 0) {
        int c = 0;
        for (int i = 0; i < S_; ++i) { c += present[i]; rank[i] = c - 1; }
    }
    __syncthreads();
    for (int p = tid; p < HW_; p += blockDim.x) {
        int r = rank[segb[p]];
        fseg_int[(size_t)b * HW_ + p] = r;
        fseg_out[(size_t)b * HW_ + p] = (float)r;
        atomicAdd(&lcnt[r], 1);
    }
    __syncthreads();
    for (int i = tid; i < S_; i += blockDim.x) cnt[b * S_ + i] = lcnt[i];
}

// ---------------------------------------------------------------------------
// Kernel 2a: pos_embed + bias -> pixel-major posb[pix][c] (LDS-tiled transpose)
// ---------------------------------------------------------------------------
__global__ void pos_bias_kernel(const float* __restrict__ pos,
                                const float* __restrict__ b1,
                                float* __restrict__ posb) {    // [HW][CI]
    __shared__ float t[CI_][CI_ + 1];
    const int p0  = blockIdx.x * 64;
    const int tid = threadIdx.x;
    for (int i = tid; i < 64 * 64; i += 256) {
        int c = i / 64, p = i % 64;                 // coalesced read along pix
        t[c][p] = pos[(size_t)c * HW_ + p0 + p] + b1[c];
    }
    __syncthreads();
    for (int i = tid; i < 64 * 64; i += 256) {
        int p = i / 64, c = i % 64;                 // coalesced write along c
        posb[(size_t)(p0 + p) * CI_ + c] = t[c][p];
    }
}

// ---------------------------------------------------------------------------
// Kernel 2b: fused conv1(7x7, 3->64, SAME) + (bias+pos) + segment-sum
// Implicit GEMM via V_WMMA_F32_16X16X4_F32; h[205MB] never materialized.
// Block: 256 thr (8 waves). Tile: 128 pixels x 64 channels; wave = 16px x 64ch.
// K-loop kept rolled (unroll 1): pinned accumulators, no register rotation.
// ---------------------------------------------------------------------------
__global__ __launch_bounds__(256)
void conv1_pool_kernel(const float* __restrict__ x,
                       const float* __restrict__ w1,
                       const float* __restrict__ posb,
                       const int* __restrict__ fseg_int,
                       float* __restrict__ sums) {             // [B][S][CI]
    __shared__ float As[MB_][KKP_];   // im2col panel, 75.8 KB
    __shared__ float Bs[CI_][KKP_];   // weights, K-contiguous per column, 37.9 KB
    const int b    = blockIdx.y;
    const int base = blockIdx.x * MB_; // HW_ % MB_ == 0
    const int tid  = threadIdx.x;

    // Stage A: im2col of 128 pixels x 147 K (zero-pad borders and K=147)
    for (int idx = tid; idx < MB_ * KKP_; idx += 256) {
        int pl = idx / KKP_, k = idx % KKP_;
        float v = 0.f;
        if (k < KK_) {
            int ci = k / 49, r = k % 49, ky = r / 7, kx = r % 7;
            int pix = base + pl;
            int y = pix / W_, xx = pix % W_;
            int gy = y + ky - 3, gx = xx + kx - 3;
            if ((unsigned)gy < (unsigned)H_ && (unsigned)gx < (unsigned)W_)
                v = x[(((size_t)b * CIN_ + ci) * H_ + gy) * W_ + gx];
        }
        As[pl][k] = v;
    }
    // Stage B: weights as [Cout][K] so fragments are contiguous b64 pairs
    for (int idx = tid; idx < CI_ * KKP_; idx += 256) {
        int co = idx / KKP_, k = idx % KKP_;
        float v = 0.f;
        if (k < KK_) {
            int ci = k / 49, r = k % 49, ky = r / 7, kx = r % 7;
            v = w1[((co * CIN_ + ci) * 7 + ky) * 7 + kx];
        }
        Bs[co][k] = v;
    }
    __syncthreads();

    const int lane  = tid & 31;
    const int wv    = tid >> 5;          // mtile 0..7
    const int ln    = lane & 15;
    const int khalf = (lane >> 4) * 2;   // ISA 16x4 f32 layout: K-half by lane[4]
    const int m     = wv * 16 + ln;

    v8f acc[4] = {};                     // 4 N-tiles of 16 channels
    #pragma unroll 1
    for (int kb = 0; kb < KKP_; kb += 4) {   // 37 K-steps, 4 WMMA each
        v2f a = *(const v2f*)&As[m][kb + khalf];        // one ds_load_b64
        #pragma unroll
        for (int j = 0; j < 4; ++j) {
            v2f bb = *(const v2f*)&Bs[j * 16 + ln][kb + khalf];
            acc[j] = __builtin_amdgcn_wmma_f32_16x16x4_f32(false, a, false, bb,
                                                           (short)0, acc[j], false, false);
        }
    }

    // Epilogue: + (bias+pos) [coalesced], then atomic segment-sum (L2-resident)
    for (int r = 0; r < 8; ++r) {
        int ml  = wv * 16 + r + (lane >> 4) * 8;   // C/D layout: M = r + 8*lane[4]
        int pix = base + ml;
        int rk  = fseg_int[(size_t)b * HW_ + pix];
        const float* pb = posb + (size_t)pix * CI_;
        float* sr = sums + ((size_t)b * S_ + rk) * CI_;
        #pragma unroll
        for (int j = 0; j < 4; ++j) {
            int c = j * 16 + ln;
            atomicAdd(&sr[c], acc[j][r] + pb[c]);
        }
    }
}

// ---------------------------------------------------------------------------
// Kernel 3a: conv2_w [768][64][3] -> n-major Bt[e][k = t*64+c]
// ---------------------------------------------------------------------------
__global__ void w2_transpose_kernel(const float* __restrict__ w2,
                                    float* __restrict__ Bt) {  // [CE][K2]
    int idx = blockIdx.x * blockDim.x + threadIdx.x;
    if (idx >= CE_ * K2_) return;
    int e = idx / K2_, k = idx % K2_;
    int t = k / CI_, c = k % CI_;
    Bt[idx] = w2[(size_t)e * K2_ + c * 3 + t];
}

// ---------------------------------------------------------------------------
// Kernel 3b: divide segment sums by clamped counts -> means (in place)
// ---------------------------------------------------------------------------
__global__ void mean_kernel(float* __restrict__ sums, const int* __restrict__ cnt) {
    int idx = blockIdx.x * blockDim.x + threadIdx.x;
    if (idx >= B_ * S_ * CI_) return;
    float c = (float)cnt[idx / CI_];
    sums[idx] = sums[idx] / fmaxf(c, 1.f);
}

// ---------------------------------------------------------------------------
// Kernel 4: conv2 (1-D, k=3, 64->768) as per-batch GEMM M=196,N=768,K=192.
// ---------------------------------------------------------------------------
__global__ __launch_bounds__(256)
void conv2_kernel(const float* __restrict__ mean,  // [B][S][CI]
                  const float* __restrict__ Bt,    // [CE][K2]
                  const float* __restrict__ b2,
                  float* __restrict__ out) {       // [B][S][CE]
    __shared__ float Am[16][K2P_];
    const int b   = blockIdx.y;
    const int s0  = blockIdx.x * 16;
    const int tid = threadIdx.x;
    for (int idx = tid; idx < 16 * K2_; idx += 256) {
        int mrow = idx / K2_, k = idx % K2_;
        int t = k / CI_, c = k % CI_;
        int s = s0 + mrow + t - 1;                 // padding=1
        float v = 0.f;
        if ((unsigned)s < (unsigned)S_) v = mean[((size_t)b * S_ + s) * CI_ + c];
        Am[mrow][k] = v;
    }
    __syncthreads();

    const int lane  = tid & 31;
    const int wv    = tid >> 5;
    const int ln    = lane & 15;
    const int khalf = (lane >> 4) * 2;

    v8f acc[6] = {};                               // 48 N-tiles / 8 waves = 6
    #pragma unroll 1
    for (int kb = 0; kb < K2_; kb += 4) {          // 48 K-steps
        v2f a = *(const v2f*)&Am[ln][kb + khalf];
        #pragma unroll
        for (int j = 0; j < 6; ++j) {
            int n = (wv + j * 8) * 16 + ln;
            v2f bb = *(const v2f*)&Bt[(size_t)n * K2_ + kb + khalf]; // global b64
            acc[j] = __builtin_amdgcn_wmma_f32_16x16x4_f32(false, a, false, bb,
                                                           (short)0, acc[j], false, false);
        }
    }

    for (int r = 0; r < 8; ++r) {
        int s = s0 + r + (lane >> 4) * 8;
        if (s < S_) {
            #pragma unroll
            for (int j = 0; j < 6; ++j) {
                int e = (wv + j * 8) * 16 + ln;
                out[((size_t)b * S_ + s) * CE_ + e] = acc[j][r] + b2[e];
            }
        }
    }
}

// ---------------------------------------------------------------------------
extern "C" void kernel_launch(void* const* d_in, const int* in_sizes, int n_in,
                              void* d_out, int out_size, void* d_ws, size_t ws_size,
                              hipStream_t stream) {
    const float* x   = (const float*)d_in[0];
    const int*   seg = (const int*)  d_in[1];
    const float* w1  = (const float*)d_in[2];
    const float* b1  = (const float*)d_in[3];
    const float* pos = (const float*)d_in[4];
    const float* w2  = (const float*)d_in[5];
    const float* b2  = (const float*)d_in[6];

    float* out      = (float*)d_out;                 // [B][S][CE]
    float* fseg_out = out + (size_t)B_ * S_ * CE_;   // [B][H][W]

    // workspace layout (~17.5 MB, 16B-aligned sections)
    char*  ws       = (char*)d_ws;
    size_t o = 0;
    int*   fseg_int = (int*)(ws + o);   o += (size_t)B_ * HW_ * 4;       // 3,211,264
    int*   cnt      = (int*)(ws + o);   o += (size_t)B_ * S_ * 4;        //    12,544
    float* sums     = (float*)(ws + o); o += (size_t)B_ * S_ * CI_ * 4;  //   802,816
    float* Bt       = (float*)(ws + o); o += (size_t)CE_ * K2_ * 4;      //   589,824
    float* posb     = (float*)(ws + o); o += (size_t)HW_ * CI_ * 4;      // 12,845,056

    // zero the atomic accumulation table (graph-capturable)
    hipMemsetAsync(sums, 0, (size_t)B_ * S_ * CI_ * sizeof(float), stream);

    seg_remap_kernel<<<B_, 256, 0, stream>>>(seg, fseg_int, fseg_out, cnt);
    pos_bias_kernel<<<HW_ / 64, 256, 0, stream>>>(pos, b1, posb);
    w2_transpose_kernel<<<(CE_ * K2_ + 255) / 256, 256, 0, stream>>>(w2, Bt);
    conv1_pool_kernel<<<dim3(HW_ / MB_, B_), 256, 0, stream>>>(x, w1, posb,
                                                               fseg_int, sums);
    mean_kernel<<<(B_ * S_ * CI_ + 255) / 256, 256, 0, stream>>>(sums, cnt);
    conv2_kernel<<<dim3((S_ + 15) / 16, B_), 256, 0, stream>>>(sums, Bt, b2, out);
}